// Bayesian_43344809951241
// MI455X (gfx1250) — compile-verified
//
#include <hip/hip_runtime.h>

// Bayesian per-example linear layer:
//   out[b,o] = sum_i x[b,i] * (mean[b,i,o] + W[b,i,o]*softplus(log_std[b,i,o])) + bias[o]
//
// Roofline: W/mean/log_std = 3 x 512^3 f32 = 1.61 GB streamed once -> ~69 us
// floor @ 23.3 TB/s; ~0.6 FLOP/byte => pure bandwidth-bound streaming kernel.
//   - every weight byte read exactly once, coalesced, non-temporal (1.6 GB
//     stream must not thrash the 192 MB L2)
//   - reduction over i via V_WMMA_F32_16X16X4_F32 in full f32 precision
//     (A = x chunk replicated across rows, B = 4x16 W_sample tile)
//   - softplus on raw hw transcendentals (v_exp_f32/v_log_f32 in base-2) to
//     keep VALU time (~20 us) well under the 69 us memory floor.

typedef float v2f __attribute__((ext_vector_type(2)));
typedef float v8f __attribute__((ext_vector_type(8)));

#define BATCH 512
#define D_IN  512
#define D_OUT 512
#define OTILES_PER_B (D_OUT / 16)          // 32 tiles of 16 outputs
#define WAVES_PER_BLOCK 8
#define THREADS (WAVES_PER_BLOCK * 32)     // wave32

__device__ __forceinline__ float softplus_f(float z) {
  // softplus(z) = max(z,0) + ln2 * log2(1 + 2^(-|z|*log2e))
  // stable for all z; v_exp_f32/v_log_f32 are ~1 ULP, plenty for f32 output.
  const float LOG2E = 1.4426950408889634f;
  const float LN2   = 0.6931471805599453f;
  float t = __builtin_amdgcn_exp2f(-fabsf(z) * LOG2E);
  return fmaxf(z, 0.0f) + LN2 * __builtin_amdgcn_logf(1.0f + t);
}

__global__ __launch_bounds__(THREADS, 1) void bayes_matvec_wmma(
    const float* __restrict__ x, const float* __restrict__ W,
    const float* __restrict__ mean, const float* __restrict__ log_std,
    const float* __restrict__ bias, float* __restrict__ out)
{
  const int lane = threadIdx.x & 31;
  // wave-uniform tile id -> SGPRs, so x loads become scalar (K$) loads
  const int tile = __builtin_amdgcn_readfirstlane(
      (int)blockIdx.x * WAVES_PER_BLOCK + ((int)threadIdx.x >> 5));
  const int b  = tile >> 5;                 // tile / OTILES_PER_B
  const int o0 = (tile & 31) << 4;          // 16-output tile base

  const int hi = lane >> 4;                 // half-wave select
  const int n  = lane & 15;                 // output column within tile

  const size_t slab = (size_t)D_IN * D_OUT;
  const float* __restrict__ Wb = W       + (size_t)b * slab;
  const float* __restrict__ Mb = mean    + (size_t)b * slab;
  const float* __restrict__ Sb = log_std + (size_t)b * slab;
  const float* __restrict__ xr = x       + (size_t)b * D_IN;

  v8f acc = {0.f, 0.f, 0.f, 0.f, 0.f, 0.f, 0.f, 0.f};

  for (int i0 = 0; i0 < D_IN; i0 += 4) {
    // wave-uniform x chunk (compiler merges to s_load_b128)
    const float x0 = xr[i0 + 0];
    const float x1 = xr[i0 + 1];
    const float x2 = xr[i0 + 2];
    const float x3 = xr[i0 + 3];

    // A (16x4 f32, rows replicated with x):
    //   lanes 0-15:  VGPR0=K0, VGPR1=K1 ; lanes 16-31: VGPR0=K2, VGPR1=K3
    v2f A;
    A.x = hi ? x2 : x0;
    A.y = hi ? x3 : x1;

    // B (4x16 f32) tile of W_sample: row k = i0 + 2*hi + {0,1}, col = o0 + n.
    // Layout assumed symmetric to documented A layout (VGPR0=K{0,2}, VGPR1=K{1,3}).
    const size_t off = (size_t)(i0 + 2 * hi) * D_OUT + (size_t)(o0 + n);
    const float w0 = __builtin_nontemporal_load(Wb + off);
    const float w1 = __builtin_nontemporal_load(Wb + off + D_OUT);
    const float m0 = __builtin_nontemporal_load(Mb + off);
    const float m1 = __builtin_nontemporal_load(Mb + off + D_OUT);
    const float s0 = __builtin_nontemporal_load(Sb + off);
    const float s1 = __builtin_nontemporal_load(Sb + off + D_OUT);

    v2f Bm;
    Bm.x = fmaf(w0, softplus_f(s0), m0);
    Bm.y = fmaf(w1, softplus_f(s1), m1);

    // acc += A x Bm   (8 args: neg_a, A, neg_b, B, c_mod, C, reuse_a, reuse_b)
    acc = __builtin_amdgcn_wmma_f32_16x16x4_f32(
        false, A, false, Bm, (short)0, acc, false, false);
  }

  // A's rows are identical, so every C row holds the same 16 dots.
  // acc[0]: lanes 0-15 => (M=0, N=lane); store the tile once from lanes 0-15.
  if (lane < 16) {
    const float r = acc[0] + bias[o0 + lane];
    __builtin_nontemporal_store(r, out + (size_t)b * D_OUT + (size_t)(o0 + lane));
  }
}

extern "C" void kernel_launch(void* const* d_in, const int* in_sizes, int n_in,
                              void* d_out, int out_size, void* d_ws, size_t ws_size,
                              hipStream_t stream) {
  const float* x       = (const float*)d_in[0];
  const float* W       = (const float*)d_in[1];
  const float* mean    = (const float*)d_in[2];
  const float* log_std = (const float*)d_in[3];
  const float* bias    = (const float*)d_in[4];
  float* out = (float*)d_out;

  const int tiles  = BATCH * OTILES_PER_B;        // 16384 waves
  const int blocks = tiles / WAVES_PER_BLOCK;     // 2048 blocks x 256 threads
  bayes_matvec_wmma<<<blocks, THREADS, 0, stream>>>(x, W, mean, log_std, bias, out);
  (void)in_sizes; (void)n_in; (void)out_size; (void)d_ws; (void)ws_size;
}